// GradientsLeastSquares_14053132992774
// MI455X (gfx1250) — compile-verified
//
#include <hip/hip_runtime.h>

// ---------------------------------------------------------------------------
// GradientsLeastSquares on gfx1250 (MI455X, wave32).
//
// Two-pass weighted LSQ gradient over a fixed-degree CSR graph.
//   pass1: grad(u)          -> first[N,3]  (stored in d_ws)
//   pass2: grad(first[:,c]) -> Hessian, emit [u_xx,u_yy,u_zz,u_xy,u_xz,u_yz]
//
// Design (see analysis): latency-bound sparse gather, fully L2-resident on
// a 192MB L2 part -> one thread per node, scalar f32 VALU, deg-16 unroll.
// CDNA5-specific paths used:
//   * global_load_async_to_lds_b128 / s_wait_asynccnt: DMA each thread's
//     contiguous 64B index row into LDS (ASYNCcnt-tracked async path).
//   * global_prefetch_b8 via __builtin_prefetch on gather targets.
// WMMA deliberately not used: edge contraction needs K=16 in f32; CDNA5 f32
// WMMA is 16x16x4 (K=4) and f16 K=32 shapes lose the 1/|dx|^2 weights'
// precision while wasting >85% of MACs on block-diagonal padding.
// ---------------------------------------------------------------------------

#define BLK    256
#define MAXDEG 16
#define EPS_W  1e-12f
#define EPS_A  1e-8f

// ---- CDNA5 async global->LDS copy (ASYNCcnt) ------------------------------
__device__ __forceinline__ void async_load_b128_to_lds(const void* gptr, void* ldsptr) {
  // Generic LDS pointer: low 32 bits are the wave-relative LDS byte offset.
  unsigned       lds_off = (unsigned)(unsigned long long)ldsptr;
  unsigned long long ga  = (unsigned long long)gptr;
  asm volatile("global_load_async_to_lds_b128 %0, %1, off"
               :: "v"(lds_off), "v"(ga)
               : "memory");
}

__device__ __forceinline__ void wait_async_all() {
  asm volatile("s_wait_asynccnt 0" ::: "memory");
}

// ---- stage this thread's index row (<=16 ints, 16B-aligned) into LDS ------
__device__ __forceinline__ int stage_indices(const int* __restrict__ offsets,
                                             const int* __restrict__ indices,
                                             int n, int* s_idx) {
  const int t = threadIdx.x;
  const int i = blockIdx.x * BLK + t;
  int deg = 0;
  if (i < n) {
    const int off0 = offsets[i];
    const int off1 = offsets[i + 1];
    deg = off1 - off0;
    if (deg > MAXDEG) deg = MAXDEG;
    int*       dst = &s_idx[t * MAXDEG];
    const int* src = &indices[off0];          // 64B-aligned (DEG=16 CSR rows)
#pragma unroll
    for (int c4 = 0; c4 < MAXDEG / 4; ++c4) {
      if (c4 * 4 < deg) async_load_b128_to_lds(src + c4 * 4, dst + c4 * 4);
    }
  }
  wait_async_all();   // each lane only reads back its own staged row
  return deg;
}

// ---- per-node weighted LSQ gradient: G = (A + eps I)^-1 b -----------------
template <int C>
__device__ __forceinline__ void lsq_node(const float* __restrict__ coords,
                                         const float* __restrict__ vals,  // [n,C]
                                         const int* sidx, int deg,
                                         float cx, float cy, float cz,
                                         const float* vi,                 // [C]
                                         float (&G)[3][C]) {
  float a00 = 0.f, a01 = 0.f, a02 = 0.f, a11 = 0.f, a12 = 0.f, a22 = 0.f;
  float b[3][C];
#pragma unroll
  for (int p = 0; p < 3; ++p)
#pragma unroll
    for (int c = 0; c < C; ++c) b[p][c] = 0.f;

  auto body = [&](int k) {
    const int j = sidx[k];
    const float dx = coords[3 * j + 0] - cx;
    const float dy = coords[3 * j + 1] - cy;
    const float dz = coords[3 * j + 2] - cz;
    const float r2 = dx * dx + dy * dy + dz * dz;
    const float w  = 1.0f / fmaxf(r2, EPS_W);
    const float wdx = w * dx, wdy = w * dy, wdz = w * dz;
    a00 += wdx * dx; a01 += wdx * dy; a02 += wdx * dz;
    a11 += wdy * dy; a12 += wdy * dz; a22 += wdz * dz;
#pragma unroll
    for (int c = 0; c < C; ++c) {
      const float dv = vals[C * j + c] - vi[c];
      b[0][c] += wdx * dv;
      b[1][c] += wdy * dv;
      b[2][c] += wdz * dv;
    }
  };

  if (deg == MAXDEG) {
#pragma unroll
    for (int k = 0; k < MAXDEG; ++k) body(k);
  } else {
    for (int k = 0; k < deg; ++k) body(k);
  }

  a00 += EPS_A; a11 += EPS_A; a22 += EPS_A;

  // symmetric 3x3 inverse via cofactors
  const float c00 = a11 * a22 - a12 * a12;
  const float c01 = a02 * a12 - a01 * a22;
  const float c02 = a01 * a12 - a02 * a11;
  const float det = a00 * c00 + a01 * c01 + a02 * c02;
  const float id  = 1.0f / det;
  const float i00 = c00 * id, i01 = c01 * id, i02 = c02 * id;
  const float i11 = (a00 * a22 - a02 * a02) * id;
  const float i12 = (a01 * a02 - a00 * a12) * id;
  const float i22 = (a00 * a11 - a01 * a01) * id;

#pragma unroll
  for (int c = 0; c < C; ++c) {
    G[0][c] = i00 * b[0][c] + i01 * b[1][c] + i02 * b[2][c];
    G[1][c] = i01 * b[0][c] + i11 * b[1][c] + i12 * b[2][c];
    G[2][c] = i02 * b[0][c] + i12 * b[1][c] + i22 * b[2][c];
  }
}

// ---- pass 1: gradient of u -> first[N,3] ----------------------------------
__global__ void __launch_bounds__(BLK)
lsq_pass1(const float* __restrict__ coords, const float* __restrict__ u,
          const int* __restrict__ offsets, const int* __restrict__ indices,
          float* __restrict__ first, int n) {
  __shared__ int s_idx[BLK * MAXDEG];
  const int deg = stage_indices(offsets, indices, n, s_idx);
  const int i = blockIdx.x * BLK + threadIdx.x;
  if (i >= n) return;
  const int* sidx = &s_idx[threadIdx.x * MAXDEG];

  if (deg == MAXDEG) {   // warm gather targets (global_prefetch_b8)
#pragma unroll
    for (int k = 0; k < MAXDEG; ++k) __builtin_prefetch(&coords[3 * sidx[k]], 0, 0);
  }

  const float cx = coords[3 * i + 0];
  const float cy = coords[3 * i + 1];
  const float cz = coords[3 * i + 2];
  const float vi[1] = {u[i]};
  float G[3][1];
  lsq_node<1>(coords, u, sidx, deg, cx, cy, cz, vi, G);
  first[3 * i + 0] = G[0][0];
  first[3 * i + 1] = G[1][0];
  first[3 * i + 2] = G[2][0];
}

// ---- pass 2: gradient of first -> 6 Hessian components --------------------
__global__ void __launch_bounds__(BLK)
lsq_pass2(const float* __restrict__ coords, const float* __restrict__ first,
          const int* __restrict__ offsets, const int* __restrict__ indices,
          float* __restrict__ out, int n) {
  __shared__ int s_idx[BLK * MAXDEG];
  const int deg = stage_indices(offsets, indices, n, s_idx);
  const int i = blockIdx.x * BLK + threadIdx.x;
  if (i >= n) return;
  const int* sidx = &s_idx[threadIdx.x * MAXDEG];

  if (deg == MAXDEG) {
#pragma unroll
    for (int k = 0; k < MAXDEG; ++k) {
      const int j = sidx[k];
      __builtin_prefetch(&coords[3 * j], 0, 0);
      __builtin_prefetch(&first[3 * j], 0, 0);
    }
  }

  const float cx = coords[3 * i + 0];
  const float cy = coords[3 * i + 1];
  const float cz = coords[3 * i + 2];
  const float vi[3] = {first[3 * i + 0], first[3 * i + 1], first[3 * i + 2]};
  float G[3][3];
  lsq_node<3>(coords, first, sidx, deg, cx, cy, cz, vi, G);

  // second[:,p,c]: out = [G00, G11, G22, G10, G20, G21]
  float* o = out + 6 * i;
  o[0] = G[0][0];
  o[1] = G[1][1];
  o[2] = G[2][2];
  o[3] = G[1][0];
  o[4] = G[2][0];
  o[5] = G[2][1];
}

// ---------------------------------------------------------------------------
extern "C" void kernel_launch(void* const* d_in, const int* in_sizes, int n_in,
                              void* d_out, int out_size, void* d_ws, size_t ws_size,
                              hipStream_t stream) {
  const float* coords  = (const float*)d_in[0];   // [N,3] f32
  const float* u       = (const float*)d_in[1];   // [N,1] f32
  const int*   offsets = (const int*)d_in[2];     // [N+1] i32
  const int*   indices = (const int*)d_in[3];     // [E]   i32
  float*       out     = (float*)d_out;           // [N,6] f32

  const int n = in_sizes[1];                      // N (u has N elements)
  float* first = (float*)d_ws;                    // scratch: N*3 floats

  const int grid = (n + BLK - 1) / BLK;
  hipLaunchKernelGGL(lsq_pass1, dim3(grid), dim3(BLK), 0, stream,
                     coords, u, offsets, indices, first, n);
  hipLaunchKernelGGL(lsq_pass2, dim3(grid), dim3(BLK), 0, stream,
                     coords, first, offsets, indices, out, n);
}